// BackProjLayer_55482387530246
// MI455X (gfx1250) — compile-verified
//
#include <hip/hip_runtime.h>

// ---------------------------------------------------------------------------
// BackProjLayer on MI455X (gfx1250).
//   latent[b,p] = Re(d_p^H S_b d_p)        (S is PSD -> eigh path collapses)
//   x = relu(conv3(relu(conv3(latent))))
//   out_b = A diag(x_b) A^H
// GEMM stages run on v_wmma_f32_16x16x4_f32 with split real/imag f32 planes.
// D is stored transposed (p-major) so WMMA B-operands and the column
// reduction are contiguous; -Si plane replaced by negating the B operand.
// ---------------------------------------------------------------------------

#define NB   8192
#define NCH  32
#define NPX  484
#define NPAD 496   // 484 padded to 31*16, zero-filled

typedef __attribute__((ext_vector_type(2))) float v2f;
typedef __attribute__((ext_vector_type(8))) float v8f;

static __device__ __forceinline__ v8f wmma4(v2f a, v2f b, v8f c) {
  // D = A(16x4) * B(4x16) + C, f32.
  return __builtin_amdgcn_wmma_f32_16x16x4_f32(false, a, false, b, (short)0, c,
                                               false, false);
}

// --------------------------- kernel 1: prep --------------------------------
// A planes row-major [32][496]; D planes TRANSPOSED [496][32]. Zero-padded.
__global__ void __launch_bounds__(256)
prep_kernel(const double* __restrict__ A, const double* __restrict__ D,
            float* __restrict__ Ar, float* __restrict__ Ai,
            float* __restrict__ DrT, float* __restrict__ DiT) {
  int idx = blockIdx.x * blockDim.x + threadIdx.x;
  if (idx >= NCH * NPAD) return;
  // A planes: idx = r*NPAD + c
  {
    int r = idx / NPAD, c = idx - r * NPAD;
    if (c < NPX) {
      size_t src = ((size_t)r * NPX + c) * 2;
      Ar[idx] = (float)A[src];
      Ai[idx] = (float)A[src + 1];
    } else {
      Ar[idx] = 0.f; Ai[idx] = 0.f;
    }
  }
  // D transposed planes: idx = p*NCH + ch
  {
    int p = idx / NCH, ch = idx - p * NCH;
    if (p < NPX) {
      size_t src = ((size_t)ch * NPX + p) * 2;
      DrT[idx] = (float)D[src];
      DiT[idx] = (float)D[src + 1];
    } else {
      DrT[idx] = 0.f; DiT[idx] = 0.f;
    }
  }
}

// --------------------------- kernel 2: latent ------------------------------
// One block per batch, 256 threads = 8 waves. Each wave owns 16-wide p-tiles.
// Tr = Sr*Dr + Si*(-Di) ;  Ti = Sr*Di + Si*Dr
// latent[p] = sum_c ( Dr[c,p]*Tr[c,p] + Di[c,p]*Ti[c,p] )
__global__ void __launch_bounds__(256)
__attribute__((amdgpu_waves_per_eu(1)))
latent_kernel(const double* __restrict__ S,
              const float* __restrict__ DrT,  // [NPAD][NCH]
              const float* __restrict__ DiT,  // [NPAD][NCH]
              float* __restrict__ latent) {
  const int b = blockIdx.x;
  __shared__ float sSr[NCH][NCH];
  __shared__ float sSi[NCH][NCH];

  const double* Sb = S + (size_t)b * NCH * NCH * 2;
  // Prefetch next batch's S while we work on this one (one line per thread).
  if (b + 1 < NB) {
    const double* Sn = Sb + (size_t)NCH * NCH * 2;
    __builtin_prefetch(Sn + (size_t)threadIdx.x * 8, 0, 0);
  }
  for (int idx = threadIdx.x; idx < NCH * NCH; idx += blockDim.x) {
    int r = idx >> 5, c = idx & 31;
    sSr[r][c] = (float)Sb[2 * idx];
    sSi[r][c] = (float)Sb[2 * idx + 1];
  }
  __syncthreads();

  const int lane = threadIdx.x & 31;
  const int wave = threadIdx.x >> 5;
  const int m16  = lane & 15;
  const int khi  = (lane >> 4) << 1;  // 0 for lanes 0-15, 2 for lanes 16-31

  // Pre-load all nt-invariant A-operands (S tiles) into registers: 64 VGPRs.
  v2f asr0[8], asr1[8], asi0[8], asi1[8];
#pragma unroll
  for (int ks = 0; ks < 8; ++ks) {
    const int ka = ks * 4 + khi;
    asr0[ks].x = sSr[m16][ka];      asr0[ks].y = sSr[m16][ka + 1];
    asr1[ks].x = sSr[m16 + 16][ka]; asr1[ks].y = sSr[m16 + 16][ka + 1];
    asi0[ks].x = sSi[m16][ka];      asi0[ks].y = sSi[m16][ka + 1];
    asi1[ks].x = sSi[m16 + 16][ka]; asi1[ks].y = sSi[m16 + 16][ka + 1];
  }

  for (int nt = wave; nt < NPAD / 16; nt += 8) {
    const int pcol = nt * 16 + m16;
    const float* dtr = DrT + (size_t)pcol * NCH;  // contiguous over c
    const float* dti = DiT + (size_t)pcol * NCH;
    v8f tr0 = {}, tr1 = {}, ti0 = {}, ti1 = {};

#pragma unroll
    for (int ks = 0; ks < 8; ++ks) {
      const int ka = ks * 4 + khi;
      // B operands: contiguous float2 along c in transposed D planes.
      v2f bdr = *(const v2f*)(dtr + ka);
      v2f bdi = *(const v2f*)(dti + ka);
      v2f bdin; bdin.x = -bdi.x; bdin.y = -bdi.y;

      tr0 = wmma4(asr0[ks], bdr,  tr0);  // + Sr*Dr
      tr0 = wmma4(asi0[ks], bdin, tr0);  // - Si*Di
      tr1 = wmma4(asr1[ks], bdr,  tr1);
      tr1 = wmma4(asi1[ks], bdin, tr1);
      ti0 = wmma4(asr0[ks], bdi,  ti0);  // + Sr*Di
      ti0 = wmma4(asi0[ks], bdr,  ti0);  // + Si*Dr
      ti1 = wmma4(asr1[ks], bdi,  ti1);
      ti1 = wmma4(asi1[ks], bdr,  ti1);
    }

    // latent[p] = sum_c Dr[c,p]*Tr[c,p] + Di[c,p]*Ti[c,p]
    // Lane holds col p=pcol, rows rbase..rbase+7 (tile 0) and +16 (tile 1):
    // reads are contiguous 8-float runs in the transposed planes.
    float part = 0.f;
    const int rbase = (lane >> 4) * 8;
#pragma unroll
    for (int j = 0; j < 8; ++j) {
      part += dtr[rbase + j]      * tr0[j] + dti[rbase + j]      * ti0[j];
      part += dtr[rbase + 16 + j] * tr1[j] + dti[rbase + 16 + j] * ti1[j];
    }
    part += __shfl_xor(part, 16, 32);
    if (lane < 16) latent[(size_t)b * NPAD + nt * 16 + lane] = part;
  }
}

// --------------------------- kernel 3: conv --------------------------------
__global__ void __launch_bounds__(512)
conv_kernel(const float* __restrict__ latent,
            const double* __restrict__ w1, const double* __restrict__ b1,
            const double* __restrict__ w2, const double* __restrict__ b2,
            float* __restrict__ xf32, double* __restrict__ xout) {
  const int b = blockIdx.x;
  __shared__ float L[NPX + 2];
  __shared__ float Y[NPX + 2];
  const int t = threadIdx.x;  // blockDim = 512

  if (t < NPX) L[t + 1] = latent[(size_t)b * NPAD + t];
  if (t == 0) { L[0] = 0.f; L[NPX + 1] = 0.f; Y[0] = 0.f; Y[NPX + 1] = 0.f; }
  __syncthreads();

  const float w10 = (float)w1[0], w11 = (float)w1[1], w12 = (float)w1[2];
  const float w20 = (float)w2[0], w21 = (float)w2[1], w22 = (float)w2[2];
  const float c1 = (float)b1[0], c2 = (float)b2[0];

  if (t < NPX) {
    float v = w10 * L[t] + w11 * L[t + 1] + w12 * L[t + 2] + c1;
    Y[t + 1] = fmaxf(v, 0.f);
  }
  __syncthreads();
  if (t < NPX) {
    float v = w20 * Y[t] + w21 * Y[t + 1] + w22 * Y[t + 2] + c2;
    v = fmaxf(v, 0.f);
    xf32[(size_t)b * NPAD + t] = v;
    xout[(size_t)b * NPX + t] = (double)v;
  }
  if (t >= NPX && t < NPAD) xf32[(size_t)b * NPAD + t] = 0.f;  // zero pad
}

// --------------------------- kernel 4: outer -------------------------------
// out_b = A diag(x_b) A^H.   out_r = (Ar.x)Ar^T + (Ai.x)Ai^T
//                            out_i = (Ai.x)Ar^T - (Ar.x)Ai^T
__global__ void __launch_bounds__(128)
outer_kernel(const float* __restrict__ Ar, const float* __restrict__ Ai,
             const float* __restrict__ xf32, double* __restrict__ out) {
  const int b = blockIdx.x;
  __shared__ float xs[NPAD];
  for (int t = threadIdx.x; t < NPAD; t += blockDim.x)
    xs[t] = xf32[(size_t)b * NPAD + t];
  __syncthreads();

  const int lane = threadIdx.x & 31;
  const int wave = threadIdx.x >> 5;  // 0..3
  const int mi = wave >> 1, li = wave & 1;
  const int m16 = lane & 15;
  const int khi = (lane >> 4) << 1;
  const float* arM = Ar + (size_t)(mi * 16 + m16) * NPAD;  // A-operand row
  const float* aiM = Ai + (size_t)(mi * 16 + m16) * NPAD;
  const float* arN = Ar + (size_t)(li * 16 + m16) * NPAD;  // B-operand row
  const float* aiN = Ai + (size_t)(li * 16 + m16) * NPAD;

  v8f accr = {}, acci = {};
  for (int k0 = 0; k0 < NPAD; k0 += 4) {
    const int kp = k0 + khi;
    const float x0 = xs[kp], x1 = xs[kp + 1];
    const v2f arm = *(const v2f*)(arM + kp);  // contiguous b64 loads
    const v2f aim = *(const v2f*)(aiM + kp);
    const v2f bR  = *(const v2f*)(arN + kp);
    const v2f bI  = *(const v2f*)(aiN + kp);
    v2f aP, aQ, aPn;
    aP.x = arm.x * x0;  aP.y = arm.y * x1;   // P = Ar .* x
    aQ.x = aim.x * x0;  aQ.y = aim.y * x1;   // Q = Ai .* x
    aPn.x = -aP.x;      aPn.y = -aP.y;       // f32 WMMA has no operand NEG

    accr = wmma4(aP,  bR, accr);   // + P Ar^T
    accr = wmma4(aQ,  bI, accr);   // + Q Ai^T
    acci = wmma4(aQ,  bR, acci);   // + Q Ar^T
    acci = wmma4(aPn, bI, acci);   // - P Ai^T
  }

  // C/D layout: VGPR j, lane l -> row = j + 8*(l>=16), col = l%16
  const int ibase = mi * 16 + (lane >> 4) * 8;
  const int l2 = li * 16 + m16;
#pragma unroll
  for (int j = 0; j < 8; ++j) {
    const size_t idx = (((size_t)b * NCH + (ibase + j)) * NCH + l2) * 2;
    double2 v;
    v.x = (double)accr[j];
    v.y = (double)acci[j];
    *reinterpret_cast<double2*>(&out[idx]) = v;  // 16B-aligned complex store
  }
}

// ---------------------------------------------------------------------------
extern "C" void kernel_launch(void* const* d_in, const int* in_sizes, int n_in,
                              void* d_out, int out_size, void* d_ws,
                              size_t ws_size, hipStream_t stream) {
  (void)in_sizes; (void)n_in; (void)out_size; (void)ws_size;
  const double* S  = (const double*)d_in[0];  // (B,1,32,32) complex128
  const double* A  = (const double*)d_in[1];  // (32,484) complex128
  const double* D  = (const double*)d_in[2];  // (32,484) complex128
  const double* w1 = (const double*)d_in[3];
  const double* b1 = (const double*)d_in[4];
  const double* w2 = (const double*)d_in[5];
  const double* b2 = (const double*)d_in[6];

  // d_out: out (B*32*32 complex128 interleaved) then x (B*484 f64)
  double* out  = (double*)d_out;
  double* xout = out + (size_t)NB * NCH * NCH * 2;

  // Workspace layout (floats)
  const size_t plane = (size_t)NCH * NPAD;  // 15872
  float* Ar  = (float*)d_ws;
  float* Ai  = Ar + plane;
  float* DrT = Ai + plane;                   // transposed [NPAD][NCH]
  float* DiT = DrT + plane;
  float* lat = DiT + plane;                  // B * NPAD
  float* xf  = lat + (size_t)NB * NPAD;      // B * NPAD

  prep_kernel<<<(int)((plane + 255) / 256), 256, 0, stream>>>(A, D, Ar, Ai, DrT, DiT);
  latent_kernel<<<NB, 256, 0, stream>>>(S, DrT, DiT, lat);
  conv_kernel<<<NB, 512, 0, stream>>>(lat, w1, b1, w2, b2, xf, xout);
  outer_kernel<<<NB, 128, 0, stream>>>(Ar, Ai, xf, out);
}